// AttentionBlock_223338299515
// MI455X (gfx1250) — compile-verified
//
#include <hip/hip_runtime.h>
#include <math.h>

#define BB 4
#define CC 128
#define HW 4096        // 64*64
#define NH 4
#define DK 32
#define R3 384         // NH * DK * 3
#define SCALE 0.17677669529663687f   // 1/sqrt(DK)

typedef __attribute__((ext_vector_type(2))) float v2f;
typedef __attribute__((ext_vector_type(8))) float v8f;

// D(16x16 f32) = A(16x4 f32) * B(4x16 f32) + C  -- true fp32 WMMA on CDNA5
__device__ __forceinline__ v8f wmma_f32(v2f a, v2f b, v8f c) {
  return __builtin_amdgcn_wmma_f32_16x16x4_f32(false, a, false, b, (short)0, c,
                                               false, false);
}

// Intra-wave LDS RAW/WAR fence (lanes are lockstep; only DScnt must drain).
__device__ __forceinline__ void wave_lds_fence() {
  __builtin_amdgcn_wave_barrier();
#if __has_builtin(__builtin_amdgcn_s_wait_dscnt)
  __builtin_amdgcn_s_wait_dscnt(0);
#else
  asm volatile("s_wait_dscnt 0" ::: "memory");
#endif
  __builtin_amdgcn_wave_barrier();
}

// ---------------------------------------------------------------------------
// K1: qkv[b, r, n] = sum_c Wp[r,c] * x[b,c,n] + bp[r]   (q rows pre-scaled)
// Each wave computes TWO 16x16 M-tiles (rows m0, m0+16) sharing the B
// operand: per wmma -> 1 shared B pair + 1 A b64 load.
// ---------------------------------------------------------------------------
__global__ __launch_bounds__(256) void qkv_proj_kernel(
    const float* __restrict__ x, const float* __restrict__ Wp,
    const float* __restrict__ bp, float* __restrict__ qkv) {
  const int lane = threadIdx.x & 31;
  const int wave = threadIdx.x >> 5;
  const int ln = lane & 15;
  const int hi = lane >> 4;
  const int kb = hi * 2;
  const int b = blockIdx.z;
  const int m0 = blockIdx.y * 64 + (wave >> 2) * 32;
  const int n0 = blockIdx.x * 64 + (wave & 3) * 16;

  const float* xb = x + (size_t)b * CC * HW;
  const float* A0 = Wp + (size_t)(m0 + ln) * CC;
  const float* A1 = Wp + (size_t)(m0 + 16 + ln) * CC;
  v8f acc0 = {}, acc1 = {};
#pragma unroll 4
  for (int k = 0; k < CC; k += 4) {
    const v2f a0 = *(const v2f*)(A0 + k + kb);   // 8B aligned: k+kb even
    const v2f a1 = *(const v2f*)(A1 + k + kb);
    v2f bv;
    bv.x = xb[(size_t)(k + kb) * HW + n0 + ln];
    bv.y = xb[(size_t)(k + kb + 1) * HW + n0 + ln];
    acc0 = wmma_f32(a0, bv, acc0);
    acc1 = wmma_f32(a1, bv, acc1);
  }
  float* outb = qkv + (size_t)b * R3 * HW;
  // q/k/v regions are 32-row aligned; a 16-row tile is uniformly one kind.
  const float s0 = ((m0 % 96) < DK) ? SCALE : 1.f;        // scalar, no lane div
  const float s1 = (((m0 + 16) % 96) < DK) ? SCALE : 1.f;
#pragma unroll
  for (int v = 0; v < 8; ++v) {
    const int r0 = m0 + v + hi * 8;
    const int r1 = r0 + 16;
    outb[(size_t)r0 * HW + n0 + ln] = (acc0[v] + bp[r0]) * s0;
    outb[(size_t)r1 * HW + n0 + ln] = (acc1[v] + bp[r1]) * s1;
  }
}

// ---------------------------------------------------------------------------
// K2: column-softmax stats. For each (b,h,j): m[j] = max_i s[i,j],
// l[j] = sum_i exp(s[i,j]-m[j]),  s[i,j] = q_i . k_j (q pre-scaled).
// Block owns 16 columns j; the K^T tile (j-invariant over the i-loop) is
// hoisted into registers; 8 waves split the 256 i-tiles, online (m,l)
// update per lane, then shuffle + LDS tree reduce.
// ---------------------------------------------------------------------------
__global__ __launch_bounds__(256) void col_stats_kernel(
    const float* __restrict__ qkv, float* __restrict__ stat_m,
    float* __restrict__ stat_l) {
  __shared__ float smm[8][16];
  __shared__ float sml[8][16];
  const int lane = threadIdx.x & 31;
  const int wave = threadIdx.x >> 5;
  const int ln = lane & 15;
  const int hi = lane >> 4;
  const int kb = hi * 2;
  const int b = blockIdx.z, h = blockIdx.y;
  const int j0 = blockIdx.x * 16;
  const float* qb = qkv + (size_t)b * R3 * HW;
  const int qrow = h * 96;
  const int krow = h * 96 + DK;

  // Hoist B operand (K^T tile at columns j0..j0+15): 8 k-steps, 2 floats each
  v2f bk[8];
#pragma unroll
  for (int k4 = 0; k4 < 8; ++k4) {
    bk[k4].x = qb[(size_t)(krow + k4 * 4 + kb) * HW + j0 + ln];
    bk[k4].y = qb[(size_t)(krow + k4 * 4 + kb + 1) * HW + j0 + ln];
  }

  float rm = -INFINITY, rl = 0.f;
  for (int it = wave; it < HW / 16; it += 8) {
    const int i0 = it * 16;
    v8f s = {};
#pragma unroll
    for (int k4 = 0; k4 < 8; ++k4) {
      v2f a;
      a.x = qb[(size_t)(qrow + k4 * 4 + kb) * HW + i0 + ln];
      a.y = qb[(size_t)(qrow + k4 * 4 + kb + 1) * HW + i0 + ln];
      s = wmma_f32(a, bk[k4], s);
    }
    float tmax = s[0];
#pragma unroll
    for (int v = 1; v < 8; ++v) tmax = fmaxf(tmax, s[v]);
    const float nm = fmaxf(rm, tmax);
    float se = 0.f;
#pragma unroll
    for (int v = 0; v < 8; ++v) se += __expf(s[v] - nm);
    rl = rl * __expf(rm - nm) + se;
    rm = nm;
  }
  // merge half-wave row groups (lane <-> lane^16 hold the same column j)
  const float om = __shfl_xor(rm, 16, 32);
  const float ol = __shfl_xor(rl, 16, 32);
  const float nm = fmaxf(rm, om);
  rl = rl * __expf(rm - nm) + ol * __expf(om - nm);
  rm = nm;
  if (hi == 0) { smm[wave][ln] = rm; sml[wave][ln] = rl; }
  __syncthreads();
  if (threadIdx.x < 16) {
    float m = -INFINITY, l = 0.f;
#pragma unroll
    for (int w = 0; w < 8; ++w) {
      const float cm = smm[w][threadIdx.x], cl = sml[w][threadIdx.x];
      const float n2 = fmaxf(m, cm);
      l = l * __expf(m - n2) + cl * __expf(cm - n2);
      m = n2;
    }
    const size_t idx = (size_t)(b * NH + h) * HW + j0 + threadIdx.x;
    stat_m[idx] = m;
    stat_l[idx] = l;
  }
}

// ---------------------------------------------------------------------------
// K3: O[i,d] = sum_j exp(s[i,j]-m[j])/l[j] * V[j,d].
// Each wave owns one 16-row i-tile, streams all 256 j-tiles: recompute S
// (8 WMMA, Q strip hoisted), softmax in-register, bounce P through LDS to
// A-layout (even row stride -> single ds_load_b64), then two P@V
// accumulations (d = 0..15, 16..31) with 4 WMMA each.
// ---------------------------------------------------------------------------
__global__ __launch_bounds__(256) void attn_out_kernel(
    const float* __restrict__ qkv, const float* __restrict__ stat_m,
    const float* __restrict__ stat_l, float* __restrict__ oattn) {
  __shared__ float pbuf[8][16][18];  // stride 18 floats: even -> b64 loads
  const int lane = threadIdx.x & 31;
  const int wave = threadIdx.x >> 5;
  const int ln = lane & 15;
  const int hi = lane >> 4;
  const int kb = hi * 2;
  const int b = blockIdx.z, h = blockIdx.y;
  const int i0 = (blockIdx.x * 8 + wave) * 16;
  const float* qb = qkv + (size_t)b * R3 * HW;
  const int qrow = h * 96;
  const int krow = h * 96 + DK;
  const int vrow = h * 96 + 2 * DK;
  const float* mj = stat_m + (size_t)(b * NH + h) * HW;
  const float* lj = stat_l + (size_t)(b * NH + h) * HW;
  const float* vr0 = qb + (size_t)(vrow + ln) * HW;        // V^T row bases
  const float* vr1 = qb + (size_t)(vrow + 16 + ln) * HW;

  // Q strip for this i-tile is invariant over j: hoist (16 floats/lane).
  v2f aq[8];
#pragma unroll
  for (int k4 = 0; k4 < 8; ++k4) {
    aq[k4].x = qb[(size_t)(qrow + k4 * 4 + kb) * HW + i0 + ln];
    aq[k4].y = qb[(size_t)(qrow + k4 * 4 + kb + 1) * HW + i0 + ln];
  }

  v8f o0 = {}, o1 = {};
  for (int jt = 0; jt < HW / 16; ++jt) {
    const int j0 = jt * 16;
    v8f s = {};
#pragma unroll
    for (int k4 = 0; k4 < 8; ++k4) {
      v2f bv;
      bv.x = qb[(size_t)(krow + k4 * 4 + kb) * HW + j0 + ln];
      bv.y = qb[(size_t)(krow + k4 * 4 + kb + 1) * HW + j0 + ln];
      s = wmma_f32(aq[k4], bv, s);
    }
    const float m = mj[j0 + ln];
    const float inv = 1.f / lj[j0 + ln];
#pragma unroll
    for (int v = 0; v < 8; ++v)                       // D-layout -> LDS
      pbuf[wave][v + hi * 8][ln] = __expf(s[v] - m) * inv;
    wave_lds_fence();
#pragma unroll
    for (int kk = 0; kk < 16; kk += 4) {              // LDS -> A-layout
      const v2f pa = *(const v2f*)(&pbuf[wave][ln][kk + kb]);  // 8B aligned
      const v2f vb0 = *(const v2f*)(vr0 + j0 + kk + kb);       // j0+kk+kb even
      const v2f vb1 = *(const v2f*)(vr1 + j0 + kk + kb);
      o0 = wmma_f32(pa, vb0, o0);
      o1 = wmma_f32(pa, vb1, o1);
    }
    wave_lds_fence();  // before next iteration overwrites pbuf
  }
  // Store O as (B, 128, N): row = h*32+d, col = i  (feeds K4 as B operand)
  float* ob = oattn + (size_t)b * (NH * DK) * HW;
#pragma unroll
  for (int v = 0; v < 8; ++v) {
    ob[(size_t)(h * DK + ln) * HW + i0 + v + hi * 8] = o0[v];
    ob[(size_t)(h * DK + 16 + ln) * HW + i0 + v + hi * 8] = o1[v];
  }
}

// ---------------------------------------------------------------------------
// K4: out[b,c,n] = sum_r Wo[c,r] * O[b,r,n] + bo[c] + x[b,c,n]
// Dual M-tile per wave, shared B operand (same scheme as K1).
// ---------------------------------------------------------------------------
__global__ __launch_bounds__(256) void out_proj_kernel(
    const float* __restrict__ oattn, const float* __restrict__ Wo,
    const float* __restrict__ bo, const float* __restrict__ x,
    float* __restrict__ out) {
  const int lane = threadIdx.x & 31;
  const int wave = threadIdx.x >> 5;
  const int ln = lane & 15;
  const int hi = lane >> 4;
  const int kb = hi * 2;
  const int b = blockIdx.z;
  const int m0 = blockIdx.y * 64 + (wave >> 2) * 32;
  const int n0 = blockIdx.x * 64 + (wave & 3) * 16;
  const float* ob = oattn + (size_t)b * CC * HW;
  const float* A0 = Wo + (size_t)(m0 + ln) * CC;
  const float* A1 = Wo + (size_t)(m0 + 16 + ln) * CC;
  v8f acc0 = {}, acc1 = {};
#pragma unroll 4
  for (int k = 0; k < CC; k += 4) {
    const v2f a0 = *(const v2f*)(A0 + k + kb);
    const v2f a1 = *(const v2f*)(A1 + k + kb);
    v2f bv;
    bv.x = ob[(size_t)(k + kb) * HW + n0 + ln];
    bv.y = ob[(size_t)(k + kb + 1) * HW + n0 + ln];
    acc0 = wmma_f32(a0, bv, acc0);
    acc1 = wmma_f32(a1, bv, acc1);
  }
#pragma unroll
  for (int v = 0; v < 8; ++v) {
    const int c0 = m0 + v + hi * 8;
    const int c1 = c0 + 16;
    const size_t i0 = ((size_t)b * CC + c0) * HW + n0 + ln;
    const size_t i1 = ((size_t)b * CC + c1) * HW + n0 + ln;
    out[i0] = acc0[v] + bo[c0] + x[i0];
    out[i1] = acc1[v] + bo[c1] + x[i1];
  }
}

// ---------------------------------------------------------------------------
extern "C" void kernel_launch(void* const* d_in, const int* in_sizes, int n_in,
                              void* d_out, int out_size, void* d_ws,
                              size_t ws_size, hipStream_t stream) {
  const float* x  = (const float*)d_in[0];
  const float* Wp = (const float*)d_in[1];
  const float* bp = (const float*)d_in[2];
  const float* Wo = (const float*)d_in[3];
  const float* bo = (const float*)d_in[4];
  float* out = (float*)d_out;

  // Workspace layout (floats): qkv 4*384*4096 | m 4*4*4096 | l 4*4*4096 |
  // oattn 4*128*4096  -> ~33.7 MB total (fits L2; attn matrix never hits HBM)
  float* qkv    = (float*)d_ws;
  float* stat_m = qkv + (size_t)BB * R3 * HW;
  float* stat_l = stat_m + (size_t)BB * NH * HW;
  float* oattn  = stat_l + (size_t)BB * NH * HW;

  qkv_proj_kernel<<<dim3(HW / 64, R3 / 64, BB), 256, 0, stream>>>(x, Wp, bp,
                                                                  qkv);
  col_stats_kernel<<<dim3(HW / 16, NH, BB), 256, 0, stream>>>(qkv, stat_m,
                                                              stat_l);
  attn_out_kernel<<<dim3(HW / 128, NH, BB), 256, 0, stream>>>(qkv, stat_m,
                                                              stat_l, oattn);
  out_proj_kernel<<<dim3(HW / 64, CC / 64, BB), 256, 0, stream>>>(oattn, Wo, bo,
                                                                  x, out);
}